// GatedLinear_26345329394132
// MI455X (gfx1250) — compile-verified
//
#include <hip/hip_runtime.h>

typedef __attribute__((ext_vector_type(2))) float v2f;
typedef __attribute__((ext_vector_type(4))) float v4f;
typedef __attribute__((ext_vector_type(8))) float v8f;

// D = A(16x4) * B(4x16) + C, fp32 WMMA (CDNA5)
#define WMMA_F32X4(a, b, c) \
  __builtin_amdgcn_wmma_f32_16x16x4_f32(false, (a), false, (b), (short)0, (c), false, false)

// ---------------------------------------------------------------------------
// Kernel 1 (templated on weight leading dim so all load offsets are
// immediates): out[:, col_base:col_base+LDN] = (x @ w) / sqrt(512)
//   x: [B,512], w: [512,LDN], out: [B,384]
// Each wave computes a 32(M) x 16(N) tile: B fragment reused by 2 WMMAs.
// ---------------------------------------------------------------------------
template <int LDN>
__global__ __launch_bounds__(256) void gl_gemm0e(
    const float* __restrict__ x,
    const float* __restrict__ w,
    float* __restrict__ out,
    int col_base, int total_tiles)
{
    const int lane = threadIdx.x & 31;
    const int wid  = threadIdx.x >> 5;
    const int tile = blockIdx.x * 8 + wid;
    if (tile >= total_tiles) return;           // whole-wave uniform exit

    constexpr int NT = LDN / 16;               // column tiles
    const int nt = tile % NT;
    const int m0 = (tile / NT) * 32;           // 2 M-tiles per wave
    const int n0 = nt * 16;

    const int half = lane >> 4;                // K sub-pair select
    const int r    = lane & 15;                // M row (A) / N col (B)

    const float* aptr0 = x + (size_t)(m0 + r) * 512 + half * 2;
    const float* aptr1 = aptr0 + 16 * 512;
    const float* bptr  = w + (size_t)(half * 2) * LDN + n0 + r;

    v8f acc0 = {}, acc1 = {};
    #pragma unroll 4
    for (int k0 = 0; k0 < 512; k0 += 4) {
        v2f b;
        b.x = bptr[k0 * LDN];                  // immediate offsets (LDN const)
        b.y = bptr[(k0 + 1) * LDN];
        v2f a0 = *(const v2f*)(aptr0 + k0);    // global_load_b64, coalesced
        v2f a1 = *(const v2f*)(aptr1 + k0);
        acc0 = WMMA_F32X4(a0, b, acc0);
        acc1 = WMMA_F32X4(a1, b, acc1);
    }

    const float scale = 0.044194173824159216f; // 1/sqrt(512)
    float* optr0 = out + (size_t)(m0 + half * 8) * 384 + col_base + n0 + r;
    float* optr1 = optr0 + 16 * 384;
    #pragma unroll
    for (int v = 0; v < 8; ++v) {
        optr0[v * 384] = acc0[v] * scale;
        optr1[v * 384] = acc1[v] * scale;
    }
}

// ---------------------------------------------------------------------------
// Kernel 2: out_1o[b,o,i] = sum_n x_1o[b,n,i] * w_1o[n,o] / sqrt(256)
//   x: [B,256,3] interleaved. One B fragment drives 3 accumulators; the
//   6 contiguous per-lane A floats come from 3 x b64 loads.
// ---------------------------------------------------------------------------
__global__ __launch_bounds__(256) void gl_gemm1o(
    const float* __restrict__ x,      // [B,256,3]
    const float* __restrict__ w,      // [256,128]
    float* __restrict__ out,          // [B,128,3]
    int total_tiles)
{
    const int lane = threadIdx.x & 31;
    const int wid  = threadIdx.x >> 5;
    const int tile = blockIdx.x * 8 + wid;
    if (tile >= total_tiles) return;

    const int NT = 8;                          // 128/16
    const int nt = tile % NT;
    const int m0 = (tile / NT) * 16;
    const int n0 = nt * 16;
    const int half = lane >> 4;
    const int r    = lane & 15;

    const float* aptr = x + ((size_t)(m0 + r) * 256 + half * 2) * 3;
    const float* bptr = w + (size_t)(half * 2) * 128 + n0 + r;

    v8f acc0 = {}, acc1 = {}, acc2 = {};
    #pragma unroll 2
    for (int k0 = 0; k0 < 256; k0 += 4) {
        v2f b;
        b.x = bptr[k0 * 128];                  // immediate offsets
        b.y = bptr[(k0 + 1) * 128];
        const float* ak = aptr + k0 * 3;       // 6 contiguous floats, 8B aligned
        v2f p0 = *(const v2f*)(ak);            // ak[0],ak[1]
        v2f p1 = *(const v2f*)(ak + 2);        // ak[2],ak[3]
        v2f p2 = *(const v2f*)(ak + 4);        // ak[4],ak[5]
        v2f a0 = {p0.x, p1.y};                 // (ak[0], ak[3])
        v2f a1 = {p0.y, p2.x};                 // (ak[1], ak[4])
        v2f a2 = {p1.x, p2.y};                 // (ak[2], ak[5])
        acc0 = WMMA_F32X4(a0, b, acc0);
        acc1 = WMMA_F32X4(a1, b, acc1);
        acc2 = WMMA_F32X4(a2, b, acc2);
    }

    const float scale = 0.0625f;               // 1/sqrt(256)
    float* optr = out + ((size_t)(m0 + half * 8) * 128 + n0 + r) * 3;
    #pragma unroll
    for (int v = 0; v < 8; ++v) {
        float* o = optr + v * 384;             // 128*3 per output row
        o[0] = acc0[v] * scale;
        o[1] = acc1[v] * scale;
        o[2] = acc2[v] * scale;
    }
}

// ---------------------------------------------------------------------------
// Kernel 3: out_2e[b,o,i] = sum_n x_2e[b,n,i] * w_2e[n,o] / sqrt(128)
//   x: [B,128,5] interleaved. One B fragment drives 5 accumulators; the
//   10 contiguous per-lane A floats (8B aligned) come from b128+b128+b64.
// ---------------------------------------------------------------------------
__global__ __launch_bounds__(256) void gl_gemm2e(
    const float* __restrict__ x,      // [B,128,5]
    const float* __restrict__ w,      // [128,64]
    float* __restrict__ out,          // [B,64,5]
    int total_tiles)
{
    const int lane = threadIdx.x & 31;
    const int wid  = threadIdx.x >> 5;
    const int tile = blockIdx.x * 8 + wid;
    if (tile >= total_tiles) return;

    const int NT = 4;                          // 64/16
    const int nt = tile % NT;
    const int m0 = (tile / NT) * 16;
    const int n0 = nt * 16;
    const int half = lane >> 4;
    const int r    = lane & 15;

    const float* aptr = x + ((size_t)(m0 + r) * 128 + half * 2) * 5;
    const float* bptr = w + (size_t)(half * 2) * 64 + n0 + r;

    v8f acc0 = {}, acc1 = {}, acc2 = {}, acc3 = {}, acc4 = {};
    #pragma unroll 2
    for (int k0 = 0; k0 < 128; k0 += 4) {
        v2f b;
        b.x = bptr[k0 * 64];                   // immediate offsets
        b.y = bptr[(k0 + 1) * 64];
        const float* ak = aptr + k0 * 5;       // 10 contiguous floats, 8B aligned
        v4f q0 = *(const v4f*)(ak);            // ak[0..3]
        v4f q1 = *(const v4f*)(ak + 4);        // ak[4..7]
        v2f q2 = *(const v2f*)(ak + 8);        // ak[8..9]
        v2f a0 = {q0.x, q1.y};                 // (ak[0], ak[5])
        v2f a1 = {q0.y, q1.z};                 // (ak[1], ak[6])
        v2f a2 = {q0.z, q1.w};                 // (ak[2], ak[7])
        v2f a3 = {q0.w, q2.x};                 // (ak[3], ak[8])
        v2f a4 = {q1.x, q2.y};                 // (ak[4], ak[9])
        acc0 = WMMA_F32X4(a0, b, acc0);
        acc1 = WMMA_F32X4(a1, b, acc1);
        acc2 = WMMA_F32X4(a2, b, acc2);
        acc3 = WMMA_F32X4(a3, b, acc3);
        acc4 = WMMA_F32X4(a4, b, acc4);
    }

    const float scale = 0.08838834764831845f;  // 1/sqrt(128)
    float* optr = out + ((size_t)(m0 + half * 8) * 64 + n0 + r) * 5;
    #pragma unroll
    for (int v = 0; v < 8; ++v) {
        float* o = optr + v * 320;             // 64*5 per output row
        o[0] = acc0[v] * scale;
        o[1] = acc1[v] * scale;
        o[2] = acc2[v] * scale;
        o[3] = acc3[v] * scale;
        o[4] = acc4[v] * scale;
    }
}

// ---------------------------------------------------------------------------
extern "C" void kernel_launch(void* const* d_in, const int* in_sizes, int n_in,
                              void* d_out, int out_size, void* d_ws, size_t ws_size,
                              hipStream_t stream) {
    const float* x0  = (const float*)d_in[0];  // [B,512,1]
    const float* x1  = (const float*)d_in[1];  // [B,256,3]
    const float* x2  = (const float*)d_in[2];  // [B,128,5]
    const float* w0s = (const float*)d_in[3];  // [512,256]
    const float* w0g = (const float*)d_in[4];  // [512,128]
    const float* w1  = (const float*)d_in[5];  // [256,128]
    const float* w2  = (const float*)d_in[6];  // [128,64]
    float* out = (float*)d_out;

    const int Bn = in_sizes[0] / 512;          // batch (100000: /16 and /32 exact)
    const int mtiles16 = Bn / 16;
    const int mtiles32 = Bn / 32;

    float* out0 = out;                                   // [B,384]
    float* out1 = out0 + (size_t)Bn * 384;               // [B,128,3]
    float* out2 = out1 + (size_t)Bn * 384;               // [B,64,5]

    const int t0s = mtiles32 * 16;   // 32x16 wave-tiles, scalar columns
    const int t0g = mtiles32 * 8;    // 32x16 wave-tiles, gate columns
    const int t1  = mtiles16 * 8;
    const int t2  = mtiles16 * 4;

    dim3 blk(256);                   // 8 wave32s; one tile per wave
    gl_gemm0e<256><<<dim3((t0s + 7) / 8), blk, 0, stream>>>(x0, w0s, out0, 0,   t0s);
    gl_gemm0e<128><<<dim3((t0g + 7) / 8), blk, 0, stream>>>(x0, w0g, out0, 256, t0g);
    gl_gemm1o<<<dim3((t1 + 7) / 8), blk, 0, stream>>>(x1, w1, out1, t1);
    gl_gemm2e<<<dim3((t2 + 7) / 8), blk, 0, stream>>>(x2, w2, out2, t2);
}